// StackedEncoder_11828339933449
// MI455X (gfx1250) — compile-verified
//
#include <hip/hip_runtime.h>
#include <cmath>

#define NN 50000          // nodes
#define NE 800000         // edges
#define FDIM 128          // feature dim
#define ZDIM 256          // 2F (concat)
#define KDIM 512          // [xh | agg] combined K

typedef float v2f __attribute__((ext_vector_type(2)));
typedef float v8f __attribute__((ext_vector_type(8)));

__device__ __forceinline__ float sigmoidf(float x) { return 1.0f / (1.0f + __expf(-x)); }

// ---------------------------------------------------------------------------
// Pack per-layer weights: Wru[512,256] = [[ws_r|ws_u];[wn_r|wn_u]],
// Wc[512,128] = [ws_c; wn_c], plus biases.
// ---------------------------------------------------------------------------
__global__ void pack_weights(const float* __restrict__ Wself, const float* __restrict__ Wneigh,
                             const float* __restrict__ bias, int layer,
                             float* __restrict__ Wru, float* __restrict__ bru,
                             float* __restrict__ Wc, float* __restrict__ bc) {
    int t = blockIdx.x * blockDim.x + threadIdx.x;
    if (t < KDIM * ZDIM) {                       // Wru
        int k = t >> 8, j = t & 255;
        int g = j >> 7, jj = j & 127;
        float v = (k < ZDIM / 2 + 128)           // k<256 -> self, else neigh
                      ? 0.0f : 0.0f;
        if (k < 256) v = Wself [(size_t)((layer * 3 + g) * 256 + k)       * 128 + jj];
        else         v = Wneigh[(size_t)((layer * 3 + g) * 256 + (k-256)) * 128 + jj];
        Wru[t] = v;
        return;
    }
    int t2 = t - KDIM * ZDIM;
    if (t2 >= 0 && t2 < KDIM * FDIM) {           // Wc
        int k = t2 >> 7, j = t2 & 127;
        float v;
        if (k < 256) v = Wself [(size_t)((layer * 3 + 2) * 256 + k)       * 128 + j];
        else         v = Wneigh[(size_t)((layer * 3 + 2) * 256 + (k-256)) * 128 + j];
        Wc[t2] = v;
        return;
    }
    int t3 = t2 - KDIM * FDIM;
    if (t3 >= 0 && t3 < ZDIM) { bru[t3] = bias[(layer * 3 + (t3 >> 7)) * 128 + (t3 & 127)]; return; }
    int t4 = t3 - ZDIM;
    if (t4 >= 0 && t4 < FDIM) { bc[t4] = bias[(layer * 3 + 2) * 128 + t4]; }
}

// ---------------------------------------------------------------------------
// xh = concat(x, h)  [N,256], float4 vectorized (N*64 float4s)
// ---------------------------------------------------------------------------
__global__ void concat_xh(const float* __restrict__ x, const float* __restrict__ h,
                          float* __restrict__ xh) {
    int t = blockIdx.x * blockDim.x + threadIdx.x;
    int row = t >> 6, q = t & 63;
    if (row >= NN) return;
    float4 v = (q < 32) ? ((const float4*)x)[(size_t)row * 32 + q]
                        : ((const float4*)h)[(size_t)row * 32 + (q - 32)];
    ((float4*)xh)[t] = v;
}

// xh2 = concat(x, r*h); r = P[:, 0:128]
__global__ void build_xh2(const float* __restrict__ x, const float* __restrict__ h,
                          const float* __restrict__ P, float* __restrict__ xh) {
    int t = blockIdx.x * blockDim.x + threadIdx.x;
    int row = t >> 6, q = t & 63;
    if (row >= NN) return;
    float4 v;
    if (q < 32) {
        v = ((const float4*)x)[(size_t)row * 32 + q];
    } else {
        int j = q - 32;
        float4 r  = ((const float4*)P)[(size_t)row * 64 + j];   // P stride 256 floats
        float4 hh = ((const float4*)h)[(size_t)row * 32 + j];
        v = make_float4(r.x * hh.x, r.y * hh.y, r.z * hh.z, r.w * hh.w);
    }
    ((float4*)xh)[t] = v;
}

__global__ void zero_f4(float4* __restrict__ p, int n4) {
    int t = blockIdx.x * blockDim.x + threadIdx.x;
    if (t < n4) p[t] = make_float4(0.f, 0.f, 0.f, 0.f);
}

// ---------------------------------------------------------------------------
// Edge scatter: agg[dst[e]] += edge_w[e] * z[src[e]].  One wave32 per edge;
// each lane handles 8 floats (2 float4 gathers, L2-resident z), hardware
// fp32 atomics into agg.
// ---------------------------------------------------------------------------
__global__ void scatter_edges(const float* __restrict__ z, const int* __restrict__ src,
                              const int* __restrict__ dst, const float* __restrict__ ew,
                              float* __restrict__ agg) {
    int gid = blockIdx.x * blockDim.x + threadIdx.x;
    int e = gid >> 5;
    int lane = threadIdx.x & 31;
    if (e >= NE) return;
    int s = src[e];
    int d = dst[e];
    float w = ew[e];
    const float4* zs = (const float4*)(z + (size_t)s * ZDIM);
    float* ad = agg + (size_t)d * ZDIM;
#pragma unroll
    for (int i = 0; i < 2; ++i) {
        float4 v = zs[lane + 32 * i];
        int b = (lane + 32 * i) * 4;
        unsafeAtomicAdd(ad + b + 0, v.x * w);
        unsafeAtomicAdd(ad + b + 1, v.y * w);
        unsafeAtomicAdd(ad + b + 2, v.z * w);
        unsafeAtomicAdd(ad + b + 3, v.w * w);
    }
}

// ---------------------------------------------------------------------------
// WMMA fp32 GEMM core: one wave owns a 16x16 output tile, K=512 split as
// xh(0:256) then agg(0:256).  B panel (512x16) staged in LDS per workgroup
// (8 waves share it).  V_WMMA_F32_16X16X4_F32, 128 matrix ops per tile.
// A frag (16x4 f32): lanes 0-15 -> M=lane, K=k0,k0+1; lanes 16-31 -> K=k0+2,k0+3.
// ---------------------------------------------------------------------------
__device__ __forceinline__ v8f wmma_tile(const float* __restrict__ rowXh,
                                         const float* __restrict__ rowAg,
                                         const float* __restrict__ Bs,
                                         int nl, int khalf) {
    v8f acc = {};
#pragma unroll 8
    for (int k0 = 0; k0 < 256; k0 += 4) {
        int kk = k0 + khalf;
        v2f a; a.x = rowXh[kk];            a.y = rowXh[kk + 1];
        v2f b; b.x = Bs[kk * 16 + nl];     b.y = Bs[(kk + 1) * 16 + nl];
        acc = __builtin_amdgcn_wmma_f32_16x16x4_f32(false, a, false, b, (short)0, acc, false, false);
    }
#pragma unroll 8
    for (int k0 = 0; k0 < 256; k0 += 4) {
        int kk = k0 + khalf;
        v2f a; a.x = rowAg[kk];                  a.y = rowAg[kk + 1];
        v2f b; b.x = Bs[(256 + kk) * 16 + nl];   b.y = Bs[(257 + kk) * 16 + nl];
        acc = __builtin_amdgcn_wmma_f32_16x16x4_f32(false, a, false, b, (short)0, acc, false, false);
    }
    return acc;
}

// Gates r,u fused: P[N,256] = sigmoid([xh|agg] @ Wru + bru)
__global__ void __launch_bounds__(256) gemm_gates_ru(const float* __restrict__ xh,
                                                     const float* __restrict__ agg,
                                                     const float* __restrict__ Wru,
                                                     const float* __restrict__ bru,
                                                     float* __restrict__ P) {
    __shared__ float Bs[KDIM * 16];              // 32 KB B panel
    const int ct = blockIdx.x;                   // 0..15 column tile
    const int tid = threadIdx.x;
    for (int i = tid; i < KDIM * 4; i += 256) {  // cooperative float4 panel load
        int k = i >> 2, q = i & 3;
        ((float4*)Bs)[i] = ((const float4*)(Wru + (size_t)k * ZDIM + ct * 16))[q];
    }
    __syncthreads();

    const int wave = tid >> 5, lane = tid & 31;
    const int rt = blockIdx.y * 8 + wave;        // row tile
    if (rt >= NN / 16) return;
    const int nl = lane & 15;
    const int khalf = (lane >> 4) << 1;
    const int m = rt * 16 + nl;

    v8f acc = wmma_tile(xh + (size_t)m * ZDIM, agg + (size_t)m * ZDIM, Bs, nl, khalf);

    const int n = ct * 16 + nl;
    const float bn = bru[n];
    const int mbase = rt * 16 + ((lane >= 16) ? 8 : 0);
#pragma unroll
    for (int r = 0; r < 8; ++r)
        P[(size_t)(mbase + r) * ZDIM + n] = sigmoidf(acc[r] + bn);
}

// Gate c + GRU combine: h_new = u*h + (1-u)*sigmoid([xh2|agg] @ Wc + bc)
__global__ void __launch_bounds__(256) gemm_gate_c(const float* __restrict__ xh2,
                                                   const float* __restrict__ agg,
                                                   const float* __restrict__ Wc,
                                                   const float* __restrict__ bc,
                                                   const float* __restrict__ P,
                                                   const float* __restrict__ hprev,
                                                   float* __restrict__ out1,
                                                   float* __restrict__ out2) {
    __shared__ float Bs[KDIM * 16];
    const int ct = blockIdx.x;                   // 0..7
    const int tid = threadIdx.x;
    for (int i = tid; i < KDIM * 4; i += 256) {
        int k = i >> 2, q = i & 3;
        ((float4*)Bs)[i] = ((const float4*)(Wc + (size_t)k * FDIM + ct * 16))[q];
    }
    __syncthreads();

    const int wave = tid >> 5, lane = tid & 31;
    const int rt = blockIdx.y * 8 + wave;
    if (rt >= NN / 16) return;
    const int nl = lane & 15;
    const int khalf = (lane >> 4) << 1;
    const int m = rt * 16 + nl;

    v8f acc = wmma_tile(xh2 + (size_t)m * ZDIM, agg + (size_t)m * ZDIM, Bs, nl, khalf);

    const int n = ct * 16 + nl;
    const float bn = bc[n];
    const int mbase = rt * 16 + ((lane >= 16) ? 8 : 0);
#pragma unroll
    for (int r = 0; r < 8; ++r) {
        int mr = mbase + r;
        float c = sigmoidf(acc[r] + bn);
        float u = P[(size_t)mr * ZDIM + FDIM + n];   // u gate from fused pass
        float hp = hprev[(size_t)mr * FDIM + n];
        float o = u * hp + (1.0f - u) * c;
        out1[(size_t)mr * FDIM + n] = o;
        if (out2) out2[(size_t)mr * FDIM + n] = o;
    }
}

// ---------------------------------------------------------------------------
extern "C" void kernel_launch(void* const* d_in, const int* in_sizes, int n_in,
                              void* d_out, int out_size, void* d_ws, size_t ws_size,
                              hipStream_t stream) {
    const float* x      = (const float*)d_in[0];
    const float* hidden = (const float*)d_in[1];
    const int*   src    = (const int*)d_in[2];
    const int*   dst    = (const int*)d_in[3];
    const float* ew     = (const float*)d_in[4];
    const float* Wself  = (const float*)d_in[5];
    const float* Wneigh = (const float*)d_in[6];
    const float* bias   = (const float*)d_in[7];
    float* out = (float*)d_out;   // [x_final | h1 | h2], each N*128

    float* ws  = (float*)d_ws;
    float* xh  = ws;                               // N*256
    float* agg = xh  + (size_t)NN * ZDIM;          // N*256
    float* P   = agg + (size_t)NN * ZDIM;          // N*256 (r|u gates)
    float* Wru = P   + (size_t)NN * ZDIM;          // 512*256
    float* bru = Wru + (size_t)KDIM * ZDIM;        // 256
    float* Wc  = bru + ZDIM;                       // 512*128
    float* bc  = Wc  + (size_t)KDIM * FDIM;        // 128

    const int n4 = NN * ZDIM / 4;                  // 3.2M float4
    dim3 blk(256);
    const int packN = (KDIM * ZDIM + KDIM * FDIM + ZDIM + FDIM + 255) / 256;
    const int rtGroups = (NN / 16 + 7) / 8;        // 391

    for (int layer = 0; layer < 2; ++layer) {
        const float* xin = (layer == 0) ? x : (out + (size_t)NN * FDIM);   // h1 feeds layer 1
        const float* hp  = hidden + (size_t)layer * NN * FDIM;
        float* o1 = out + (size_t)(1 + layer) * NN * FDIM;                 // hiddens[layer]
        float* o2 = (layer == 1) ? out : nullptr;                          // x_final == h2

        pack_weights<<<packN, blk, 0, stream>>>(Wself, Wneigh, bias, layer, Wru, bru, Wc, bc);

        // gconv for gates r,u (shared agg)
        concat_xh<<<NN * 64 / 256, blk, 0, stream>>>(xin, hp, xh);
        zero_f4<<<(n4 + 255) / 256, blk, 0, stream>>>((float4*)agg, n4);
        scatter_edges<<<(NE * 32 + 255) / 256, blk, 0, stream>>>(xh, src, dst, ew, agg);
        gemm_gates_ru<<<dim3(16, rtGroups), blk, 0, stream>>>(xh, agg, Wru, bru, P);

        // gconv for gate c with xh2 = [x, r*h]
        build_xh2<<<NN * 64 / 256, blk, 0, stream>>>(xin, hp, P, xh);
        zero_f4<<<(n4 + 255) / 256, blk, 0, stream>>>((float4*)agg, n4);
        scatter_edges<<<(NE * 32 + 255) / 256, blk, 0, stream>>>(xh, src, dst, ew, agg);
        gemm_gate_c<<<dim3(8, rtGroups), blk, 0, stream>>>(xh, agg, Wc, bc, P, hp, o1, o2);
    }
}